// JaCDE_88167088653055
// MI455X (gfx1250) — compile-verified
//
#include <hip/hip_runtime.h>

// B=8192, NOBS=16, CIN=64, H=128
// out[b] = jx + jxh + jxhh where F(y) = dtanh ⊙ ((drelu ⊙ (y @ wh^T)) @ wout^T)
// One 16-row batch tile per block; 2 waves per block each own 64 output columns.

typedef __attribute__((ext_vector_type(2))) float v2f;
typedef __attribute__((ext_vector_type(8))) float v8f;

#define LDP   132   // padded LDS row stride (floats): 132 % 64 == 4 -> conflict-free frag reads
#define HDIM  128
#define CINC  64
#define NOBSC 16

__device__ __forceinline__ v8f wmma4(v2f a, v2f b, v8f c) {
  // D(16x16,f32) = A(16x4,f32) * B(4x16,f32) + C
  return __builtin_amdgcn_wmma_f32_16x16x4_f32(
      /*neg_a=*/false, a, /*neg_b=*/false, b,
      /*c_mod=*/(short)0, c, /*reuse_a=*/false, /*reuse_b=*/false);
}

// acc[nt] += Y(16xK, in LDS row-major, stride LDP) @ W(128xK row-major)^T, n-tiles at wcol..wcol+63
template <int K>
__device__ __forceinline__ void gemm_lds(const float* __restrict__ sY,
                                         const float* __restrict__ W, int ldw,
                                         int wcol, int lane, v8f acc[4]) {
  const int mrow  = lane & 15;          // A row (M) and B column-within-tile (N)
  const int khalf = (lane >> 4) << 1;   // lanes 0-15 hold K={0,1}, lanes 16-31 K={2,3}
  #pragma unroll 4
  for (int ks = 0; ks < K / 4; ++ks) {
    const int k0 = ks * 4 + khalf;
    v2f a = *(const v2f*)(sY + mrow * LDP + k0);
    #pragma unroll
    for (int nt = 0; nt < 4; ++nt) {
      const int n = wcol + nt * 16 + mrow;
      v2f b = *(const v2f*)(W + n * ldw + k0);   // B[k][n] = W[n][k]
      acc[nt] = wmma4(a, b, acc[nt]);
    }
  }
}

// scatter 4 C/D fragments (16x16 each) into LDS row-major at columns wcol..wcol+63
__device__ __forceinline__ void write_tile(float* __restrict__ sY, int wcol, int lane,
                                           const v8f v[4]) {
  const int hi   = lane >> 4;
  const int col0 = lane & 15;
  #pragma unroll
  for (int nt = 0; nt < 4; ++nt) {
    const int col = wcol + nt * 16 + col0;
    #pragma unroll
    for (int r = 0; r < 8; ++r)
      sY[(r + 8 * hi) * LDP + col] = v[nt][r];
  }
}

__global__ void __launch_bounds__(64) jacde_kernel(
    const float* __restrict__ t,      const float* __restrict__ hmat,
    const float* __restrict__ coeffs, const float* __restrict__ dcoeffs,
    const float* __restrict__ tobs,
    const float* __restrict__ wx,     const float* __restrict__ wh,
    const float* __restrict__ wout,
    const float* __restrict__ b0,     const float* __restrict__ b1,
    float* __restrict__ out) {
  __shared__ float sY[16 * LDP];

  const int tid  = threadIdx.x;
  const int lane = tid & 31;
  const int wave = tid >> 5;
  const int wcol = wave * 64;            // this wave's 64 output columns
  const int brow = blockIdx.x * 16;      // 16 batch rows per block
  const int hi   = lane >> 4;
  const int col0 = lane & 15;

  // ---- scalar spline setup (searchsorted right over 16-entry tobs) ----
  const float ts = t[0];
  int idx = 0;
  #pragma unroll
  for (int j = 0; j < NOBSC; ++j) idx += (tobs[j] <= ts) ? 1 : 0;
  idx -= 1;
  idx = idx < 0 ? 0 : (idx > NOBSC - 2 ? NOBSC - 2 : idx);
  const float dt = ts - tobs[idx];
  const float p1 = dt, p2 = dt * dt, p3 = p2 * dt;

  // ---- stage x = cubic(coeffs) into sY[:,0:64]  (thread tid -> channel, i -> row) ----
  {
    const float* cb = coeffs + ((size_t)brow * (NOBSC - 1) + idx) * CINC * 4;
    #pragma unroll 4
    for (int i = 0; i < 16; ++i) {
      const float4 c4 = *(const float4*)(cb + (size_t)i * (NOBSC - 1) * CINC * 4 + tid * 4);
      sY[i * LDP + tid] = c4.x + c4.y * p1 + c4.z * p2 + c4.w * p3;
    }
  }
  __syncthreads();

  // ---- l1 = x @ wx^T + h @ wh^T + b0 ----
  v8f acc[4];
  #pragma unroll
  for (int nt = 0; nt < 4; ++nt) {
    const float bv = b0[wcol + nt * 16 + col0];
    #pragma unroll
    for (int r = 0; r < 8; ++r) acc[nt][r] = bv;
  }
  gemm_lds<CINC>(sY, wx, CINC, wcol, lane, acc);
  __syncthreads();

  #pragma unroll 4
  for (int i = 0; i < 32; ++i) {               // stage h tile (16x128)
    const int v = i * 64 + tid, m = v >> 7, c = v & 127;
    sY[m * LDP + c] = hmat[(size_t)(brow + m) * HDIM + c];
  }
  __syncthreads();
  gemm_lds<HDIM>(sY, wh, HDIM, wcol, lane, acc);
  __syncthreads();

  // ---- drelu = sigmoid(l1); relu = max(l1,0) ----
  v8f drelu[4];
  #pragma unroll
  for (int nt = 0; nt < 4; ++nt)
    #pragma unroll
    for (int r = 0; r < 8; ++r) {
      const float x = acc[nt][r];
      drelu[nt][r] = 1.0f / (1.0f + __expf(-x));
      acc[nt][r]   = fmaxf(x, 0.0f);
    }
  write_tile(sY, wcol, lane, acc);             // relu -> LDS
  __syncthreads();

  // ---- lout = relu @ wout^T + b1 ; dtanh = 1 - tanh^2 ----
  v8f dtn[4];
  {
    v8f a2[4];
    #pragma unroll
    for (int nt = 0; nt < 4; ++nt) {
      const float bv = b1[wcol + nt * 16 + col0];
      #pragma unroll
      for (int r = 0; r < 8; ++r) a2[nt][r] = bv;
    }
    gemm_lds<HDIM>(sY, wout, HDIM, wcol, lane, a2);
    #pragma unroll
    for (int nt = 0; nt < 4; ++nt)
      #pragma unroll
      for (int r = 0; r < 8; ++r) {
        const float th = tanhf(a2[nt][r]);
        dtn[nt][r] = 1.0f - th * th;
      }
  }
  __syncthreads();

  // ---- stage xdot, u = xdot @ wx^T ----
  {
    const float* db = dcoeffs + ((size_t)brow * (NOBSC - 1) + idx) * CINC * 4;
    #pragma unroll 4
    for (int i = 0; i < 16; ++i) {
      const float4 c4 = *(const float4*)(db + (size_t)i * (NOBSC - 1) * CINC * 4 + tid * 4);
      sY[i * LDP + tid] = c4.x + c4.y * p1 + c4.z * p2 + c4.w * p3;
    }
  }
  __syncthreads();
  v8f cur[4];
  #pragma unroll
  for (int nt = 0; nt < 4; ++nt) cur[nt] = (v8f)(0.0f);
  gemm_lds<CINC>(sY, wx, CINC, wcol, lane, cur);
  __syncthreads();

  // ---- jx = dtanh ⊙ ((drelu ⊙ u) @ wout^T) ----
  #pragma unroll
  for (int nt = 0; nt < 4; ++nt) cur[nt] *= drelu[nt];
  write_tile(sY, wcol, lane, cur);
  __syncthreads();
  v8f outv[4];
  #pragma unroll
  for (int nt = 0; nt < 4; ++nt) outv[nt] = (v8f)(0.0f);
  gemm_lds<HDIM>(sY, wout, HDIM, wcol, lane, outv);
  __syncthreads();
  #pragma unroll
  for (int nt = 0; nt < 4; ++nt) {
    outv[nt] *= dtn[nt];
    cur[nt] = outv[nt];                        // cur = jx, outv accumulates result
  }

  // ---- jxh, jxhh: two applications of F ----
  for (int it = 0; it < 2; ++it) {
    write_tile(sY, wcol, lane, cur);
    __syncthreads();
    v8f t1[4];
    #pragma unroll
    for (int nt = 0; nt < 4; ++nt) t1[nt] = (v8f)(0.0f);
    gemm_lds<HDIM>(sY, wh, HDIM, wcol, lane, t1);
    __syncthreads();
    #pragma unroll
    for (int nt = 0; nt < 4; ++nt) t1[nt] *= drelu[nt];
    write_tile(sY, wcol, lane, t1);
    __syncthreads();
    v8f t2[4];
    #pragma unroll
    for (int nt = 0; nt < 4; ++nt) t2[nt] = (v8f)(0.0f);
    gemm_lds<HDIM>(sY, wout, HDIM, wcol, lane, t2);
    __syncthreads();
    #pragma unroll
    for (int nt = 0; nt < 4; ++nt) {
      t2[nt] *= dtn[nt];
      outv[nt] += t2[nt];
      cur[nt] = t2[nt];
    }
  }

  // ---- store (fragment layout -> row-major global) ----
  #pragma unroll
  for (int nt = 0; nt < 4; ++nt) {
    const int col = wcol + nt * 16 + col0;
    #pragma unroll
    for (int r = 0; r < 8; ++r)
      out[(size_t)(brow + r + 8 * hi) * HDIM + col] = outv[nt][r];
  }
}

extern "C" void kernel_launch(void* const* d_in, const int* in_sizes, int n_in,
                              void* d_out, int out_size, void* d_ws, size_t ws_size,
                              hipStream_t stream) {
  const float* t       = (const float*)d_in[0];
  const float* hmat    = (const float*)d_in[1];
  const float* coeffs  = (const float*)d_in[2];
  const float* dcoeffs = (const float*)d_in[3];
  const float* tobs    = (const float*)d_in[4];
  const float* wx      = (const float*)d_in[5];
  const float* wh      = (const float*)d_in[6];
  const float* wout    = (const float*)d_in[7];
  const float* b0      = (const float*)d_in[8];
  const float* b1      = (const float*)d_in[9];
  float* out           = (float*)d_out;

  const int B = 8192;
  jacde_kernel<<<B / 16, 64, 0, stream>>>(t, hmat, coeffs, dcoeffs, tobs,
                                          wx, wh, wout, b0, b1, out);
}